// SimpleConditionalNet_16733192585677
// MI455X (gfx1250) — compile-verified
//
#include <hip/hip_runtime.h>
#include <stdint.h>

// out = x + c, where c = 2 if any(x >= 1), else 4 if any(x >= -1), else 5.
// (Exact collapse of the reference's lax.cond layer chain.)
// Memory-bound: ~256-384 MiB HBM traffic -> ~11-16 us floor at 23.3 TB/s.
// No matrix FLOPs exist in this workload, so WMMA is inapplicable; the CDNA5
// paths exercised are the async global->LDS engine (ASYNCcnt), split wait
// counters (s_wait_asynccnt / s_wait_dscnt), wave32 reductions, and NT b128
// stores. The 128 MiB input fits the 192 MB L2, so pass 2's read mostly hits L2.

typedef float v4f __attribute__((ext_vector_type(4)));

#define THREADS 256
#define P1_BLOCKS 2048
#define P2_BLOCKS 4096

__global__ void zero_flags_kernel(uint32_t* __restrict__ f) {
    f[0] = 0u;
}

// Pass 1: global any(x>=1) -> bit0, any(x>=-1) -> bit1 of flags[0].
// Streams x through LDS with a 2-deep async_load_to_lds pipeline; each lane
// stages 2 x b128 per stage (1 KB per wave per stage). LDS is partitioned per
// lane so each lane reads back exactly the bytes its own wave's async
// instructions wrote: s_wait_asynccnt alone orders producer -> consumer, no
// workgroup barrier in the steady-state loop.
__global__ __launch_bounds__(THREADS)
void pass1_flags_kernel(const float* __restrict__ x,
                        uint32_t* __restrict__ flags, int n)
{
    __shared__ __align__(16) float smem[2][THREADS * 8]; // 2 x 8 KiB staging
    __shared__ uint32_t sflag;
    const int tid = threadIdx.x;
    if (tid == 0) sflag = 0u;
    __syncthreads();

    // Low 32 bits of the generic pointer to LDS == byte offset in the wave's
    // LDS allocation (flat-aperture rule) -> exactly what async VDST expects.
    const uint32_t ldsA = (uint32_t)(uintptr_t)(&smem[0][0]) + (uint32_t)(tid * 16);
    const uint32_t ldsB = (uint32_t)(uintptr_t)(&smem[1][0]) + (uint32_t)(tid * 16);
    const uint32_t HALF = THREADS * 16;                  // 4096 B: second b128 set

    const int nstages = n >> 11;            // 256 threads * 8 floats per stage
    int s = (int)blockIdx.x;
    bool a1 = false, a2 = false;

    if (s < nstages) {                      // prologue: fill buffer 0 (2 issues)
        uint32_t g0 = (uint32_t)s * (THREADS * 32) + (uint32_t)(tid * 16);
        asm volatile("global_load_async_to_lds_b128 %0, %1, %2"
                     :: "v"(ldsA), "v"(g0) , "s"(x) : "memory");
        asm volatile("global_load_async_to_lds_b128 %0, %1, %2"
                     :: "v"(ldsA + HALF), "v"(g0 + HALF), "s"(x) : "memory");
    }

    int i = 0;
    for (; s < nstages; s += (int)gridDim.x, ++i) {
        const int snext = s + (int)gridDim.x;
        if (snext < nstages) {
            const uint32_t lds_next = (i & 1) ? ldsA : ldsB;
            uint32_t g0 = (uint32_t)snext * (THREADS * 32) + (uint32_t)(tid * 16);
            // WAR guard: buffer being overwritten was ds_load-read last
            // iteration; ensure those DS reads have landed in VGPRs.
            asm volatile("s_wait_dscnt 0x0" ::: "memory");
            asm volatile("global_load_async_to_lds_b128 %0, %1, %2"
                         :: "v"(lds_next), "v"(g0), "s"(x) : "memory");
            asm volatile("global_load_async_to_lds_b128 %0, %1, %2"
                         :: "v"(lds_next + HALF), "v"(g0 + HALF), "s"(x) : "memory");
            // 4 outstanding; async loads complete in order -> <=2 means the
            // older buffer's pair has fully landed in LDS.
            asm volatile("s_wait_asynccnt 0x2" ::: "memory");
        } else {
            asm volatile("s_wait_asynccnt 0x0" ::: "memory"); // drain
        }
        const v4f v0 = *(const v4f*)(&smem[i & 1][tid * 4]);            // ds_load_b128
        const v4f v1 = *(const v4f*)(&smem[i & 1][THREADS * 4 + tid * 4]);
        a1 = a1 | (v0.x >= 1.0f)  | (v0.y >= 1.0f)  | (v0.z >= 1.0f)  | (v0.w >= 1.0f)
                | (v1.x >= 1.0f)  | (v1.y >= 1.0f)  | (v1.z >= 1.0f)  | (v1.w >= 1.0f);
        a2 = a2 | (v0.x >= -1.0f) | (v0.y >= -1.0f) | (v0.z >= -1.0f) | (v0.w >= -1.0f)
                | (v1.x >= -1.0f) | (v1.y >= -1.0f) | (v1.z >= -1.0f) | (v1.w >= -1.0f);
    }

    uint32_t f = (a1 ? 1u : 0u) | (a2 ? 2u : 0u);
    if (f) atomicOr(&sflag, f);              // LDS atomic (ds_or)
    __syncthreads();
    if (tid == 0 && sflag) atomicOr(flags, sflag); // one global atomic / block
}

// Pass 2: out = x + c. Input largely resident in the 192 MB L2 from pass 1;
// output stored non-temporally (never re-read on device -> keep L2 clean).
__global__ __launch_bounds__(THREADS)
void pass2_add_kernel(const float* __restrict__ x, float* __restrict__ out,
                      const uint32_t* __restrict__ flags, int n4)
{
    const uint32_t f = flags[0];                       // uniform s_load
    const float c = (f & 1u) ? 2.0f : ((f & 2u) ? 4.0f : 5.0f);
    const v4f* __restrict__ xi = (const v4f*)x;
    v4f* __restrict__ yo = (v4f*)out;
    int i = (int)(blockIdx.x * blockDim.x + threadIdx.x);
    const int stride = (int)(gridDim.x * blockDim.x);
#pragma unroll 4
    for (; i < n4; i += stride) {
        v4f v = xi[i];                                 // global_load_b128 (RT)
        v += c;
        __builtin_nontemporal_store(v, &yo[i]);        // global_store_b128 NT
    }
}

extern "C" void kernel_launch(void* const* d_in, const int* in_sizes, int n_in,
                              void* d_out, int out_size, void* d_ws, size_t ws_size,
                              hipStream_t stream) {
    const float* x = (const float*)d_in[0];
    float* out = (float*)d_out;
    uint32_t* flags = (uint32_t*)d_ws;
    const int n = in_sizes[0];                         // 4096 * 8192

    zero_flags_kernel<<<1, 1, 0, stream>>>(flags);
    pass1_flags_kernel<<<P1_BLOCKS, THREADS, 0, stream>>>(x, flags, n);
    pass2_add_kernel<<<P2_BLOCKS, THREADS, 0, stream>>>(x, out, flags, n / 4);
}